// BiMamba_82471962018255
// MI455X (gfx1250) — compile-verified
//
#include <hip/hip_runtime.h>
#include <hip/hip_bf16.h>

// ---------------------------------------------------------------------------
// out[b] = sum_t x[b,t,:] · (Bin · A^(T-1-t) · w)
//
// Phase 1: seed  G[:,0:16]  = [w, Aw, ..., A^15 w]        (tiny sequential)
// Phase 2: 4 squarings A^2..A^16, then 7 doubling rounds:
//             G[:, c:2c] = A^c · G[:, 0:c],  A^(2c) = (A^c)^2   (WMMA f32)
// Phase 3: P[t,:] = Bin · G[:, T-1-t]                      (WMMA f32 GEMM)
// Phase 4: out[b] = dot(x[b].flat, P.flat)                 (HBM-bound sweep)
// ---------------------------------------------------------------------------

typedef __attribute__((ext_vector_type(2))) float v2f;
typedef __attribute__((ext_vector_type(8))) float v8f;

#define D_DIM 256
#define T_DIM 2048

// One 16x16 f32 output tile: C = A(row0:,:256) * B(:256, col0:), via
// V_WMMA_F32_16X16X4_F32.  A row-major lda, B row-major ldb.
// ISA 7.12.2 layouts: A 16x4: lanes 0-15 hold K=0,1 ; lanes 16-31 K=2,3.
//                     B 4x16: mirrored (per-lane N, VGPR pair = 2 K rows).
__device__ __forceinline__ v8f wmma_tile_f32(const float* __restrict__ A, int lda,
                                             const float* __restrict__ B, int ldb,
                                             int row0, int col0) {
    v8f acc = {};
    const int lane = threadIdx.x & 31;
    const int half = lane >> 4;   // 0: K pair {0,1}, 1: K pair {2,3}
    const int l    = lane & 15;
    const float* ap = A + (size_t)(row0 + l) * lda + 2 * half;
    const float* bp = B + (size_t)(2 * half) * ldb + col0 + l;
#pragma unroll 8
    for (int k = 0; k < D_DIM; k += 4) {
        v2f a, b;
        a.x = ap[k];
        a.y = ap[k + 1];
        const float* bk = bp + (size_t)k * ldb;
        b.x = bk[0];
        b.y = bk[ldb];
        acc = __builtin_amdgcn_wmma_f32_16x16x4_f32(
            /*neg_a=*/false, a, /*neg_b=*/false, b,
            /*c_mod=*/(short)0, acc, /*reuse_a=*/false, /*reuse_b=*/false);
    }
    return acc;
}

// Store 16x16 f32 C/D tile (VGPR i: lanes0-15 -> M=i, lanes16-31 -> M=8+i).
__device__ __forceinline__ void store_tile_f32(float* __restrict__ C, int ldc,
                                               int row0, int col0, v8f acc) {
    const int lane = threadIdx.x & 31;
    const int half = lane >> 4;
    const int l    = lane & 15;
    float* cp = C + (size_t)(row0 + 8 * half) * ldc + col0 + l;
#pragma unroll
    for (int i = 0; i < 8; ++i) cp[(size_t)i * ldc] = acc[i];
}

// ---------------------------------------------------------------------------
// Phase 1: G[:, s] = A^s w for s=0..15 (sequential matvecs, one workgroup).
// G is row-major [D][T] (ld = T_DIM).
__global__ void seed_kernel(const float* __restrict__ A, const float* __restrict__ w,
                            float* __restrict__ G) {
    __shared__ float g[D_DIM];
    __shared__ float gn[D_DIM];
    const int d = threadIdx.x;
    g[d] = w[d];
    G[(size_t)d * T_DIM + 0] = g[d];
    __syncthreads();
    for (int s = 1; s < 16; ++s) {
        const float* row = A + (size_t)d * D_DIM;
        float acc = 0.f;
#pragma unroll 8
        for (int e = 0; e < D_DIM; ++e) acc += row[e] * g[e];
        gn[d] = acc;
        __syncthreads();
        g[d] = gn[d];
        G[(size_t)d * T_DIM + s] = g[d];
        __syncthreads();
    }
}

// ---------------------------------------------------------------------------
// Phase 2 unified step: waves [0,gTiles) compute G[:, cols+16c] = M*G[:,16c]
// (doubling), waves [gTiles, gTiles+256) compute Mnext = M*M (squaring).
// Branches are wave-uniform -> EXEC stays all-ones around WMMA.
__global__ void step_kernel(const float* __restrict__ M, float* __restrict__ Mnext,
                            int doSquare, float* __restrict__ G, int cols, int gTiles) {
    const int w = blockIdx.x * 8 + (threadIdx.x >> 5);
    if (w < gTiles) {
        const int tr = w & 15;   // 16 row tiles of D
        const int tc = w >> 4;   // column tiles of current G width
        v8f acc = wmma_tile_f32(M, D_DIM, G, T_DIM, tr * 16, tc * 16);
        store_tile_f32(G, T_DIM, tr * 16, cols + tc * 16, acc);
    } else if (doSquare) {
        const int t = w - gTiles;
        if (t < 256) {
            const int tr = t & 15;
            const int tc = t >> 4;
            v8f acc = wmma_tile_f32(M, D_DIM, M, D_DIM, tr * 16, tc * 16);
            store_tile_f32(Mnext, D_DIM, tr * 16, tc * 16, acc);
        }
    }
}

// ---------------------------------------------------------------------------
// Phase 3: C[d,s] = sum_e Bin[d,e] * G[e,s]; scatter-store flipped so that
// P[t, d] = C[d, T-1-t]  (row-major [T][D], contiguous in d for phase 4).
__global__ void pmat_kernel(const float* __restrict__ Bin, const float* __restrict__ G,
                            float* __restrict__ P) {
    const int w  = blockIdx.x * 8 + (threadIdx.x >> 5);  // 2048 waves total
    const int tr = w & 15;   // d tile
    const int tc = w >> 4;   // s tile (0..127)
    v8f acc = wmma_tile_f32(Bin, D_DIM, G, T_DIM, tr * 16, tc * 16);
    const int lane = threadIdx.x & 31;
    const int half = lane >> 4;
    const int l    = lane & 15;
    const int s  = tc * 16 + l;
    const int d0 = tr * 16 + 8 * half;
    float* cp = P + (size_t)(T_DIM - 1 - s) * D_DIM + d0;
#pragma unroll
    for (int i = 0; i < 8; ++i) cp[i] = acc[i];  // contiguous 8-float store
}

// ---------------------------------------------------------------------------
// Phase 4: 64 chunks per batch row; partials[b*64+chunk] = chunk dot product.
// 134 MB of x streamed once (HBM-bound); P (2 MB) re-read from L2.
__global__ void reduce_kernel(const float* __restrict__ x, const float* __restrict__ P,
                              float* __restrict__ partials) {
    const int b     = blockIdx.x >> 6;
    const int chunk = blockIdx.x & 63;
    const int per4  = (T_DIM * D_DIM / 4) / 64;   // 2048 float4 per chunk
    const float4* xp = (const float4*)(x + (size_t)b * T_DIM * D_DIM) + chunk * per4;
    const float4* pp = (const float4*)P + chunk * per4;
    float acc = 0.f;
    for (int i = threadIdx.x; i < per4; i += 256) {
        __builtin_prefetch(&xp[i + 512], 0, 1);   // global_prefetch_b8
        float4 xv = xp[i];
        float4 pv = pp[i];
        acc += xv.x * pv.x + xv.y * pv.y + xv.z * pv.z + xv.w * pv.w;
    }
    for (int off = 16; off > 0; off >>= 1) acc += __shfl_down(acc, off, 32);
    __shared__ float s[8];
    if ((threadIdx.x & 31) == 0) s[threadIdx.x >> 5] = acc;
    __syncthreads();
    if (threadIdx.x == 0) {
        float t = 0.f;
#pragma unroll
        for (int i = 0; i < 8; ++i) t += s[i];
        partials[blockIdx.x] = t;
    }
}

__global__ void finish_kernel(const float* __restrict__ partials, float* __restrict__ out) {
    const int b = blockIdx.x;
    float v = partials[b * 64 + threadIdx.x];   // 64 threads
    for (int off = 16; off > 0; off >>= 1) v += __shfl_down(v, off, 32);
    __shared__ float s2[2];
    if ((threadIdx.x & 31) == 0) s2[threadIdx.x >> 5] = v;
    __syncthreads();
    if (threadIdx.x == 0) out[b] = s2[0] + s2[1];
}

// ---------------------------------------------------------------------------
extern "C" void kernel_launch(void* const* d_in, const int* in_sizes, int n_in,
                              void* d_out, int out_size, void* d_ws, size_t ws_size,
                              hipStream_t stream) {
    const float* x   = (const float*)d_in[0];  // [64, 2048, 256]
    const float* A   = (const float*)d_in[1];  // [256, 256] state_matrix
    const float* Bin = (const float*)d_in[2];  // [256, 256] input_matrix
    const float* wv  = (const float*)d_in[3];  // [256, 1]   output_matrix
    float* out = (float*)d_out;                // [64]
    float* ws  = (float*)d_ws;

    float* M1 = ws;                        // 65536 floats
    float* M2 = ws + 65536;                // 65536 floats
    float* G  = ws + 131072;               // [256][2048] row-major
    float* P  = ws + 131072 + 524288;      // [2048][256] row-major
    float* pr = ws + 131072 + 2 * 524288;  // 4096 partials

    // Phase 1: G cols 0..15
    seed_kernel<<<1, 256, 0, stream>>>(A, wv, G);
    // Phase 2a: A^2, A^4, A^8, A^16 (pure squaring; 256 tiles = 32 blocks)
    step_kernel<<<32, 256, 0, stream>>>(A,  M1, 1, G, 0, 0);
    step_kernel<<<32, 256, 0, stream>>>(M1, M2, 1, G, 0, 0);
    step_kernel<<<32, 256, 0, stream>>>(M2, M1, 1, G, 0, 0);
    step_kernel<<<32, 256, 0, stream>>>(M1, M2, 1, G, 0, 0);   // M2 = A^16
    // Phase 2b: doubling rounds; blocks = ceil((gTiles + 256*doSquare)/8)
    step_kernel<<<34,  256, 0, stream>>>(M2, M1, 1, G, 16,   16);   // G->32,  M1=A^32
    step_kernel<<<36,  256, 0, stream>>>(M1, M2, 1, G, 32,   32);   // G->64,  M2=A^64
    step_kernel<<<40,  256, 0, stream>>>(M2, M1, 1, G, 64,   64);   // G->128, M1=A^128
    step_kernel<<<48,  256, 0, stream>>>(M1, M2, 1, G, 128,  128);  // G->256, M2=A^256
    step_kernel<<<64,  256, 0, stream>>>(M2, M1, 1, G, 256,  256);  // G->512, M1=A^512
    step_kernel<<<96,  256, 0, stream>>>(M1, M2, 1, G, 512,  512);  // G->1024,M2=A^1024
    step_kernel<<<128, 256, 0, stream>>>(M2, M1, 0, G, 1024, 1024); // G->2048
    // Phase 3: P = (Bin * G) flipped  (2048 tiles = 256 blocks)
    pmat_kernel<<<256, 256, 0, stream>>>(Bin, G, P);
    // Phase 4: streaming dot products
    reduce_kernel<<<64 * 64, 256, 0, stream>>>(x, P, pr);
    finish_kernel<<<64, 64, 0, stream>>>(pr, out);
}